// _IPEXScaleDotProductCPU_54477365182764
// MI455X (gfx1250) — compile-verified
//
#include <hip/hip_runtime.h>
#include <hip/hip_bf16.h>
#include <math.h>

// Problem sizes (fixed by the reference).
#define B_  16
#define S_  1
#define H_  32
#define D_  128
#define T_  2048
#define TK_ 2049   // T + S

typedef __attribute__((ext_vector_type(2))) float v2f;
typedef __attribute__((ext_vector_type(4))) float v4f;
typedef __attribute__((ext_vector_type(8))) float v8f;

// One workgroup per (b,h). 256 threads = 8 wave32.
// Fused: beam-gather cache copy -> kc/vc outputs, flash-decode attention from LDS.
// Streaming traffic (~2.1 GB) uses non-temporal loads/stores so the 192MB L2 is
// not thrashed by read-once / write-once data.
__global__ __launch_bounds__(256)
void mha_decode_fused(const float* __restrict__ q_in,
                      const float* __restrict__ k_in,
                      const float* __restrict__ v_in,
                      const float* __restrict__ kc_in,
                      const float* __restrict__ vc_in,
                      const float* __restrict__ mask,
                      const int*   __restrict__ beam,
                      float* __restrict__ out_attn,
                      float* __restrict__ out_kc,
                      float* __restrict__ out_vc)
{
    __shared__ float sQ[D_];
    __shared__ float sK[128][D_ + 4];   // +4 pad: WMMA A-operand gather from LDS
    __shared__ float sV[128][D_];
    __shared__ float sS[128];           // scaled+masked scores for current tile
    __shared__ float sRed[256];         // final cross-half combine

    const int bh   = blockIdx.x;
    const int b    = bh / H_;
    const int h    = bh % H_;
    const int tid  = threadIdx.x;
    const int lane = tid & 31;
    const int wave = tid >> 5;          // 0..7
    const int dOwn = tid & (D_ - 1);    // output dim owned by this thread
    const int kh   = tid >> 7;          // key-half of the tile (0: keys 0-63, 1: 64-127)
    const float scale = 0.08838834764831845f;  // 1/sqrt(128)

    const int bsrc = beam[b];
    const float* kcS = kc_in + (size_t)(bsrc * H_ + h) * T_ * D_;
    const float* vcS = vc_in + (size_t)(bsrc * H_ + h) * T_ * D_;
    float* kcD = out_kc + (size_t)(b * H_ + h) * TK_ * D_;
    float* vcD = out_vc + (size_t)(b * H_ + h) * TK_ * D_;
    const float* maskB = mask + (size_t)b * TK_;

    if (tid < D_) sQ[tid] = q_in[(size_t)(b * H_ + h) * D_ + tid];

    float m_run = -INFINITY;   // running softmax max (redundant per thread)
    float l_run = 0.0f;        // running softmax denom (redundant per thread)
    float acc   = 0.0f;        // partial out[dOwn] over this thread's key half

    // WMMA f32 16x16x4 A-operand addressing (documented layout):
    //   lanes 0-15 feed K dims {0,1} of row M=lane; lanes 16-31 feed dims {2,3} of row M=lane-16.
    const int mrowA = wave * 16 + (lane & 15);
    const int ksel  = (lane >> 4) << 1;

    for (int t0 = 0; t0 < T_; t0 += 128) {
        // ---- Stream 128-key tile once: global -> (output cache, LDS). Fully coalesced,
        //      non-temporal in both directions (read-once in, write-once out). ----
        #pragma unroll
        for (int j = 0; j < 16; ++j) {
            int idx = tid + j * 256;          // 0..4095 float4 slots (128 rows x 32 float4)
            int row = idx >> 5;
            int col = (idx & 31) << 2;
            size_t off = (size_t)(t0 + row) * D_ + col;
            v4f kvec = __builtin_nontemporal_load((const v4f*)(kcS + off));
            __builtin_nontemporal_store(kvec, (v4f*)(kcD + off));
            *(v4f*)(&sK[row][col]) = kvec;
            v4f vvec = __builtin_nontemporal_load((const v4f*)(vcS + off));
            __builtin_nontemporal_store(vvec, (v4f*)(vcD + off));
            *(v4f*)(&sV[row][col]) = vvec;
        }
        if (t0 + 128 < T_) {
            // 256 threads x 256B stride covers the entire next 64KB K and V tiles.
            __builtin_prefetch(kcS + (size_t)(t0 + 128) * D_ + tid * 64, 0, 0);
            __builtin_prefetch(vcS + (size_t)(t0 + 128) * D_ + tid * 64, 0, 0);
        }
        __syncthreads();

        // ---- Scores for this wave's 16 keys via fp32 WMMA: D = K_tile(16x4) x q(4x16) chain. ----
        // B-operand: q chunk replicated across all 16 N columns, so every D column equals the
        // score and no cross-lane extraction is needed.
        v8f accS = {};
        #pragma unroll
        for (int c = 0; c < D_; c += 4) {
            v2f a, bq;
            a.x  = sK[mrowA][c + ksel];
            a.y  = sK[mrowA][c + ksel + 1];
            bq.x = sQ[c + ksel];
            bq.y = sQ[c + ksel + 1];
            accS = __builtin_amdgcn_wmma_f32_16x16x4_f32(
                       false, a, false, bq, (short)0, accS, false, false);
        }
        // D layout: VGPR i -> M=i (lanes 0-15) / M=i+8 (lanes 16-31). Lanes 0 and 16 publish.
        if ((lane & 15) == 0) {
            int base = wave * 16 + ((lane >> 4) << 3);
            #pragma unroll
            for (int i = 0; i < 8; ++i) {
                int key = base + i;
                sS[key] = accS[i] * scale + maskB[t0 + key];
            }
        }
        __syncthreads();

        // ---- Online softmax update (identical, deterministic per thread) + PV accumulate. ----
        float m_tile = -INFINITY;
        #pragma unroll 8
        for (int i = 0; i < 128; ++i) m_tile = fmaxf(m_tile, sS[i]);
        float m_new = fmaxf(m_run, m_tile);
        float corr  = __expf(m_run - m_new);
        float lsum  = 0.0f;
        #pragma unroll 8
        for (int i = 0; i < 128; ++i) lsum += __expf(sS[i] - m_new);
        acc *= corr;
        const int kbase = kh << 6;
        #pragma unroll 8
        for (int i = 0; i < 64; ++i) {
            float p = __expf(sS[kbase + i] - m_new);
            acc = fmaf(p, sV[kbase + i][dOwn], acc);
        }
        l_run = l_run * corr + lsum;
        m_run = m_new;
        __syncthreads();
    }

    // ---- Append the new token (t = T) to kc/vc and fold it into the softmax. ----
    if (tid < D_) {
        float kn = k_in[(size_t)(b * H_ + h) * D_ + tid];
        float vn = v_in[(size_t)(b * H_ + h) * D_ + tid];
        sK[0][tid] = kn;
        sV[0][tid] = vn;
        __builtin_nontemporal_store(kn, kcD + (size_t)T_ * D_ + tid);
        __builtin_nontemporal_store(vn, vcD + (size_t)T_ * D_ + tid);
    }
    __syncthreads();
    float sNew = 0.0f;
    #pragma unroll 8
    for (int i = 0; i < 128; ++i) sNew = fmaf(sQ[i], sK[0][i], sNew);
    sNew = sNew * scale + maskB[T_];
    float m_new = fmaxf(m_run, sNew);
    float corr  = __expf(m_run - m_new);
    float p     = __expf(sNew - m_new);
    acc *= corr;
    if (kh == 0) acc = fmaf(p, sV[0][dOwn], acc);   // add new token's PV exactly once per dim
    l_run = l_run * corr + p;

    // ---- Combine the two key-halves and normalize. ----
    sRed[tid] = acc;
    __syncthreads();
    if (tid < D_) {
        float o = (sRed[tid] + sRed[tid + 128]) / l_run;
        out_attn[(size_t)(b * H_ + h) * D_ + tid] = o;   // [B,S,H,D], S==1
    }
}

extern "C" void kernel_launch(void* const* d_in, const int* in_sizes, int n_in,
                              void* d_out, int out_size, void* d_ws, size_t ws_size,
                              hipStream_t stream) {
    (void)in_sizes; (void)n_in; (void)out_size; (void)d_ws; (void)ws_size;
    const float* q    = (const float*)d_in[0];
    const float* k    = (const float*)d_in[1];
    const float* v    = (const float*)d_in[2];
    const float* kc   = (const float*)d_in[3];
    const float* vc   = (const float*)d_in[4];
    const float* mk   = (const float*)d_in[5];
    const int*   beam = (const int*)d_in[6];

    // d_out = [attn_output | kc | vc] flat in return order.
    float* out_attn = (float*)d_out;
    float* out_kc   = out_attn + (size_t)B_ * S_ * H_ * D_;
    float* out_vc   = out_kc   + (size_t)B_ * H_ * TK_ * D_;

    mha_decode_fused<<<dim3(B_ * H_), dim3(256), 0, stream>>>(
        q, k, v, kc, vc, mk, beam, out_attn, out_kc, out_vc);
}